// GCN_57741540328069
// MI455X (gfx1250) — compile-verified
//
#include <hip/hip_runtime.h>

// CDNA5 / gfx1250, wave32. Uses V_WMMA_F32_16X16X4_F32 for all dense projections.

typedef __attribute__((ext_vector_type(2))) float v2f;
typedef __attribute__((ext_vector_type(8))) float v8f;

#define N_NODES 50000
#define N_EDGES 800000
#define F_IN 128

// ---- workspace layout (float offsets) ----
// h0   : [0,              N*64)   node features after feat-linear + embeddings
// agg1 : [N*64,           2*N*64) conv1 neighbor sums
// h1   : [2*N*64,         2*N*64 + N*16)
// agg2 : [.. + N*16,      .. + 2*N*16)
// cnt  : next N floats
// inv  : next N floats
#define OFF_H0   ((size_t)0)
#define OFF_AGG1 ((size_t)N_NODES * 64)
#define OFF_H1   ((size_t)N_NODES * 128)
#define OFF_AGG2 (OFF_H1 + (size_t)N_NODES * 16)
#define OFF_CNT  (OFF_AGG2 + (size_t)N_NODES * 16)
#define OFF_INV  (OFF_CNT + (size_t)N_NODES)

static __device__ __forceinline__ v8f wmma_f32(v2f a, v2f b, v8f c) {
  return __builtin_amdgcn_wmma_f32_16x16x4_f32(false, a, false, b, (short)0, c,
                                               false, false);
}

__global__ void zero_kernel(float* __restrict__ p, int n) {
  int t = blockIdx.x * blockDim.x + threadIdx.x;
  if (t < n) p[t] = 0.0f;
}

// h0[:, 0:32] = x @ W_feat.T + b_feat      (fp32 WMMA, 16-node tile per wave)
__global__ __launch_bounds__(128) void feat_gemm(
    const float* __restrict__ x, const float* __restrict__ Wf,
    const float* __restrict__ bf, float* __restrict__ h0) {
  int tile = blockIdx.x * 4 + (threadIdx.x >> 5);
  if (tile * 16 >= N_NODES) return;           // wave-uniform exit (EXEC stays full)
  int lane = threadIdx.x & 31;
  int half = lane >> 4;                       // K-half: lanes 0-15 hold K={0,1}, 16-31 K={2,3}
  int l = lane & 15;
  int nb = tile * 16;

  v8f acc0 = {}, acc1 = {};
  const float* xrow = x + (size_t)(nb + l) * F_IN + 2 * half;   // A: row m=l
  const float* w0 = Wf + (size_t)l * F_IN + 2 * half;           // B col n=l   (W_feat[n][k])
  const float* w1 = Wf + (size_t)(16 + l) * F_IN + 2 * half;    // B col n=16+l
  for (int k0 = 0; k0 < F_IN; k0 += 4) {
    v2f a  = *(const v2f*)(xrow + k0);
    v2f b0 = *(const v2f*)(w0 + k0);
    v2f b1 = *(const v2f*)(w1 + k0);
    acc0 = wmma_f32(a, b0, acc0);
    acc1 = wmma_f32(a, b1, acc1);
  }
  float bb0 = bf[l], bb1 = bf[16 + l];
  for (int r = 0; r < 8; ++r) {
    int m = r + 8 * half;                     // C/D: VGPR r -> rows r / r+8
    h0[(size_t)(nb + m) * 64 + l]      = acc0[r] + bb0;
    h0[(size_t)(nb + m) * 64 + 16 + l] = acc1[r] + bb1;
  }
}

// h0[:, 32:48] = reg_emb[reg_id], h0[:, 48:64] = dep_emb[dep_id]
__global__ void embed_kernel(const int* __restrict__ reg_id,
                             const int* __restrict__ dep_id,
                             const float* __restrict__ reg_emb,
                             const float* __restrict__ dep_emb,
                             float* __restrict__ h0) {
  int t = blockIdx.x * blockDim.x + threadIdx.x;
  if (t >= N_NODES * 16) return;
  int node = t >> 4, f = t & 15;
  h0[(size_t)node * 64 + 32 + f] = reg_emb[reg_id[node] * 16 + f];
  h0[(size_t)node * 64 + 48 + f] = dep_emb[dep_id[node] * 16 + f];
}

__global__ void degree_kernel(const int* __restrict__ ei, float* __restrict__ cnt) {
  int e = blockIdx.x * blockDim.x + threadIdx.x;
  if (e < N_EDGES) atomicAdd(&cnt[ei[N_EDGES + e]], 1.0f);
}

__global__ void inv_kernel(const float* __restrict__ cnt, float* __restrict__ inv) {
  int i = blockIdx.x * blockDim.x + threadIdx.x;
  if (i < N_NODES) inv[i] = 1.0f / fmaxf(cnt[i], 1.0f);
}

// conv1 scatter: agg1[dst] += h0[src]  (64 floats/edge, 16 threads/edge, float4)
__global__ void gather64(const int* __restrict__ ei, const float* __restrict__ h,
                         float* __restrict__ agg) {
  int t = blockIdx.x * blockDim.x + threadIdx.x;
  if (t >= N_EDGES * 16) return;
  int e = t >> 4, c = t & 15;
  int s = ei[e], d = ei[N_EDGES + e];
  float4 v = *(const float4*)(h + (size_t)s * 64 + c * 4);
  float* o = agg + (size_t)d * 64 + c * 4;
  atomicAdd(o + 0, v.x); atomicAdd(o + 1, v.y);
  atomicAdd(o + 2, v.z); atomicAdd(o + 3, v.w);
}

// conv2 scatter: agg2[dst] += h1[src]  (16 floats/edge, 4 threads/edge)
__global__ void gather16(const int* __restrict__ ei, const float* __restrict__ h,
                         float* __restrict__ agg) {
  int t = blockIdx.x * blockDim.x + threadIdx.x;
  if (t >= N_EDGES * 4) return;
  int e = t >> 2, c = t & 3;
  int s = ei[e], d = ei[N_EDGES + e];
  float4 v = *(const float4*)(h + (size_t)s * 16 + c * 4);
  float* o = agg + (size_t)d * 16 + c * 4;
  atomicAdd(o + 0, v.x); atomicAdd(o + 1, v.y);
  atomicAdd(o + 2, v.z); atomicAdd(o + 3, v.w);
}

// h1 = relu(mean1 @ W1_l.T + b1_l + h0 @ W1_r.T), K=64, 16 outputs
__global__ __launch_bounds__(128) void combine1(
    const float* __restrict__ h0, const float* __restrict__ agg1,
    const float* __restrict__ inv, const float* __restrict__ W1l,
    const float* __restrict__ b1l, const float* __restrict__ W1r,
    float* __restrict__ h1) {
  int tile = blockIdx.x * 4 + (threadIdx.x >> 5);
  if (tile * 16 >= N_NODES) return;
  int lane = threadIdx.x & 31, half = lane >> 4, l = lane & 15, nb = tile * 16;

  v8f acc = {};
  float iv = inv[nb + l];
  const float* arow = agg1 + (size_t)(nb + l) * 64 + 2 * half;
  const float* hrow = h0 + (size_t)(nb + l) * 64 + 2 * half;
  const float* wl = W1l + (size_t)l * 64 + 2 * half;   // W1_l[n][k], n=l
  const float* wr = W1r + (size_t)l * 64 + 2 * half;
  for (int k0 = 0; k0 < 64; k0 += 4) {
    v2f a1; a1.x = arow[k0] * iv; a1.y = arow[k0 + 1] * iv;   // mean = agg * inv
    v2f b1 = *(const v2f*)(wl + k0);
    acc = wmma_f32(a1, b1, acc);
    v2f a2 = *(const v2f*)(hrow + k0);
    v2f b2 = *(const v2f*)(wr + k0);
    acc = wmma_f32(a2, b2, acc);
  }
  float bb = b1l[l];
  for (int r = 0; r < 8; ++r) {
    int m = r + 8 * half;
    h1[(size_t)(nb + m) * 16 + l] = fmaxf(acc[r] + bb, 0.0f);
  }
}

// out2 = relu(mean2 @ W2_l.T + b2_l + h1 @ W2_r.T) [N,32]; y = out2 @ W_lin.T + b_lin
__global__ __launch_bounds__(128) void combine2_final(
    const float* __restrict__ h1, const float* __restrict__ agg2,
    const float* __restrict__ inv, const float* __restrict__ W2l,
    const float* __restrict__ b2l, const float* __restrict__ W2r,
    const float* __restrict__ Wlin, const float* __restrict__ blin,
    float* __restrict__ y) {
  int tile = blockIdx.x * 4 + (threadIdx.x >> 5);
  if (tile * 16 >= N_NODES) return;
  int lane = threadIdx.x & 31, half = lane >> 4, l = lane & 15, nb = tile * 16;

  v8f acc0 = {}, acc1 = {};
  float iv = inv[nb + l];
  const float* arow = agg2 + (size_t)(nb + l) * 16 + 2 * half;
  const float* hrow = h1 + (size_t)(nb + l) * 16 + 2 * half;
  for (int k0 = 0; k0 < 16; k0 += 4) {
    v2f a1; a1.x = arow[k0] * iv; a1.y = arow[k0 + 1] * iv;
    v2f a2 = *(const v2f*)(hrow + k0);
    v2f bl0 = *(const v2f*)(W2l + (size_t)l * 16 + 2 * half + k0);
    v2f bl1 = *(const v2f*)(W2l + (size_t)(16 + l) * 16 + 2 * half + k0);
    v2f br0 = *(const v2f*)(W2r + (size_t)l * 16 + 2 * half + k0);
    v2f br1 = *(const v2f*)(W2r + (size_t)(16 + l) * 16 + 2 * half + k0);
    acc0 = wmma_f32(a1, bl0, acc0);
    acc0 = wmma_f32(a2, br0, acc0);
    acc1 = wmma_f32(a1, bl1, acc1);
    acc1 = wmma_f32(a2, br1, acc1);
  }
  float bb0 = b2l[l], bb1 = b2l[16 + l];
  float w0 = Wlin[l], w1 = Wlin[16 + l];
  float s[8];
  for (int r = 0; r < 8; ++r) {
    float v0 = fmaxf(acc0[r] + bb0, 0.0f);   // out2 column l
    float v1 = fmaxf(acc1[r] + bb1, 0.0f);   // out2 column 16+l
    s[r] = v0 * w0 + v1 * w1;                // partial dot with W_lin
  }
  // reduce across the 16 lanes of each half (halves hold disjoint row sets)
  for (int off = 1; off < 16; off <<= 1)
    for (int r = 0; r < 8; ++r) s[r] += __shfl_xor(s[r], off, 16);
  if (l == 0) {
    float bl = blin[0];
    for (int r = 0; r < 8; ++r) y[nb + r + 8 * half] = s[r] + bl;
  }
}

extern "C" void kernel_launch(void* const* d_in, const int* in_sizes, int n_in,
                              void* d_out, int out_size, void* d_ws, size_t ws_size,
                              hipStream_t stream) {
  const float* x       = (const float*)d_in[0];
  const int*   ei      = (const int*)d_in[1];     // [2, E] (src row, dst row)
  const int*   reg_id  = (const int*)d_in[2];
  const int*   dep_id  = (const int*)d_in[3];
  const float* W_feat  = (const float*)d_in[4];
  const float* b_feat  = (const float*)d_in[5];
  const float* reg_emb = (const float*)d_in[6];
  const float* dep_emb = (const float*)d_in[7];
  const float* W1_l    = (const float*)d_in[8];
  const float* b1_l    = (const float*)d_in[9];
  const float* W1_r    = (const float*)d_in[10];
  const float* W2_l    = (const float*)d_in[11];
  const float* b2_l    = (const float*)d_in[12];
  const float* W2_r    = (const float*)d_in[13];
  const float* W_lin   = (const float*)d_in[14];
  const float* b_lin   = (const float*)d_in[15];
  float* y = (float*)d_out;

  float* ws   = (float*)d_ws;
  float* h0   = ws + OFF_H0;
  float* agg1 = ws + OFF_AGG1;
  float* h1   = ws + OFF_H1;
  float* agg2 = ws + OFF_AGG2;
  float* cnt  = ws + OFF_CNT;
  float* inv  = ws + OFF_INV;

  const int TILES = (N_NODES + 15) / 16;              // 3125
  const int GBLK  = (TILES + 3) / 4;                  // 4 waves / block

  // zero agg1 and the contiguous [agg2 | cnt] region
  {
    int n1 = N_NODES * 64;
    zero_kernel<<<(n1 + 255) / 256, 256, 0, stream>>>(agg1, n1);
    int n2 = N_NODES * 17;                            // agg2 (16) + cnt (1)
    zero_kernel<<<(n2 + 255) / 256, 256, 0, stream>>>(agg2, n2);
  }

  feat_gemm<<<GBLK, 128, 0, stream>>>(x, W_feat, b_feat, h0);
  embed_kernel<<<(N_NODES * 16 + 255) / 256, 256, 0, stream>>>(reg_id, dep_id,
                                                               reg_emb, dep_emb, h0);
  degree_kernel<<<(N_EDGES + 255) / 256, 256, 0, stream>>>(ei, cnt);
  inv_kernel<<<(N_NODES + 255) / 256, 256, 0, stream>>>(cnt, inv);

  gather64<<<(N_EDGES * 16 + 255) / 256, 256, 0, stream>>>(ei, h0, agg1);
  combine1<<<GBLK, 128, 0, stream>>>(h0, agg1, inv, W1_l, b1_l, W1_r, h1);

  gather16<<<(N_EDGES * 4 + 255) / 256, 256, 0, stream>>>(ei, h1, agg2);
  combine2_final<<<GBLK, 128, 0, stream>>>(h1, agg2, inv, W2_l, b2_l, W2_r,
                                           W_lin, b_lin, y);
}